// ScaleDotAttention_24575802868290
// MI455X (gfx1250) — compile-verified
//
#include <hip/hip_runtime.h>
#include <hip/hip_bf16.h>
#include <math.h>

// Problem constants from the reference
#define BB 32
#define SS 4096
#define DD 1024   // DQ == DK == DV == 1024

typedef float v2f __attribute__((ext_vector_type(2)));
typedef float v8f __attribute__((ext_vector_type(8)));

// ---------------------------------------------------------------------------
// Kernel 1: qW = q @ W using full-precision fp32 WMMA (V_WMMA_F32_16X16X4_F32)
// grid = (B/16, DK/16) = (2, 64), block = 32 (one wave per 16x16 tile)
// A (16x4 f32) layout: lanes 0-15 / 16-31 both hold M=0..15; VGPR0/1 hold
// K = kg, kg+1 where kg = (lane>=16 ? 2 : 0).  B (4x16) is the dual layout.
// C/D: VGPR i -> row (i + (lane>=16 ? 8 : 0)), col = lane&15.
// ---------------------------------------------------------------------------
__global__ __launch_bounds__(32)
void sdpa_qw_wmma_kernel(const float* __restrict__ q,
                         const float* __restrict__ W,
                         float* __restrict__ qW) {
  const int lane = threadIdx.x;            // 0..31
  const int mn   = lane & 15;
  const int kg   = (lane >> 4) << 1;       // 0 or 2
  const int rowT = blockIdx.x << 4;        // 0 or 16
  const int colT = blockIdx.y << 4;        // 0..1008

  const float* qrow = q + (size_t)(rowT + mn) * DD;
  v8f acc = {};
  #pragma unroll 4
  for (int k0 = 0; k0 < DD; k0 += 4) {
    v2f a, b;
    a.x = qrow[k0 + kg];
    a.y = qrow[k0 + kg + 1];
    b.x = W[(size_t)(k0 + kg)     * DD + colT + mn];
    b.y = W[(size_t)(k0 + kg + 1) * DD + colT + mn];
    acc = __builtin_amdgcn_wmma_f32_16x16x4_f32(
        /*neg_a=*/false, a, /*neg_b=*/false, b,
        /*c_mod=*/(short)0, acc, /*reuse_a=*/false, /*reuse_b=*/false);
  }

  const int mBase = rowT + ((lane >> 4) << 3);
  #pragma unroll
  for (int i = 0; i < 8; ++i)
    qW[(size_t)(mBase + i) * DD + colT + mn] = acc[i];
}

// ---------------------------------------------------------------------------
// Kernel 2: score[b,s] = k[b,s,:] . qW[b,:]   (streams k once, coalesced b128)
// grid = (S/128, B), block = 256 (8 waves); each wave handles 16 rows.
// Each lane preloads its 32 qW values (8 x float4) into registers.
// ---------------------------------------------------------------------------
__global__ __launch_bounds__(256)
void sdpa_score_kernel(const float* __restrict__ k,
                       const float* __restrict__ qW,
                       float* __restrict__ score) {
  const int b    = blockIdx.y;
  const int wave = threadIdx.x >> 5;
  const int lane = threadIdx.x & 31;

  const float4* qW4 = (const float4*)(qW + (size_t)b * DD);
  float4 qr[8];
  #pragma unroll
  for (int j = 0; j < 8; ++j) qr[j] = qW4[lane + 32 * j];

  const int sBase = blockIdx.x * 128 + wave * 16;
  for (int r = 0; r < 16; ++r) {
    const int s = sBase + r;
    const float4* krow = (const float4*)(k + ((size_t)b * SS + s) * DD);
    float acc = 0.f;
    #pragma unroll
    for (int j = 0; j < 8; ++j) {
      float4 kv = krow[lane + 32 * j];
      acc += kv.x * qr[j].x;
      acc += kv.y * qr[j].y;
      acc += kv.z * qr[j].z;
      acc += kv.w * qr[j].w;
    }
    // wave32 butterfly reduction
    #pragma unroll
    for (int off = 16; off > 0; off >>= 1)
      acc += __shfl_xor(acc, off, 32);
    if (lane == 0) score[(size_t)b * SS + s] = acc;
  }
}

// ---------------------------------------------------------------------------
// Kernel 3: numerically stable softmax over S per batch; writes attn output.
// grid = B, block = 256; each thread owns 16 strided elements.
// ---------------------------------------------------------------------------
__global__ __launch_bounds__(256)
void sdpa_softmax_kernel(const float* __restrict__ score,
                         float* __restrict__ attn) {
  const int b   = blockIdx.x;
  const int tid = threadIdx.x;
  __shared__ float red[256];
  const float* sc = score + (size_t)b * SS;

  float m = -3.402823466e38f;
  #pragma unroll
  for (int i = 0; i < 16; ++i) m = fmaxf(m, sc[tid + 256 * i]);
  red[tid] = m; __syncthreads();
  for (int st = 128; st > 0; st >>= 1) {
    if (tid < st) red[tid] = fmaxf(red[tid], red[tid + st]);
    __syncthreads();
  }
  const float gmax = red[0];
  __syncthreads();

  float ssum = 0.f;
  #pragma unroll
  for (int i = 0; i < 16; ++i) ssum += expf(sc[tid + 256 * i] - gmax);
  red[tid] = ssum; __syncthreads();
  for (int st = 128; st > 0; st >>= 1) {
    if (tid < st) red[tid] += red[tid + st];
    __syncthreads();
  }
  const float inv = 1.0f / red[0];

  float* at = attn + (size_t)b * SS;
  #pragma unroll
  for (int i = 0; i < 16; ++i)
    at[tid + 256 * i] = expf(sc[tid + 256 * i] - gmax) * inv;
}

// ---------------------------------------------------------------------------
// Kernel 4: partial out[b, c, d] = sum_{s in chunk c} attn[b,s] * v[b,s,d]
// grid = (8 chunks of 512 rows, B), block = 256; thread owns a float4 of d.
// Streams v exactly once with fully coalesced float4 loads.
// ---------------------------------------------------------------------------
__global__ __launch_bounds__(256)
void sdpa_outpart_kernel(const float* __restrict__ v,
                         const float* __restrict__ attn,
                         float* __restrict__ part) {
  const int b   = blockIdx.y;   // 0..31
  const int c   = blockIdx.x;   // 0..7
  const int tid = threadIdx.x;  // 0..255  -> d = 4*tid

  __shared__ float aw[512];
  const float* at = attn + (size_t)b * SS + (size_t)c * 512;
  aw[tid]       = at[tid];
  aw[tid + 256] = at[tid + 256];
  __syncthreads();

  const float4* v4 = (const float4*)(v + ((size_t)b * SS + (size_t)c * 512) * DD);
  float4 acc = make_float4(0.f, 0.f, 0.f, 0.f);
  #pragma unroll 4
  for (int s = 0; s < 512; ++s) {
    const float  w  = aw[s];
    const float4 vv = v4[(size_t)s * 256 + tid];
    acc.x += w * vv.x;
    acc.y += w * vv.y;
    acc.z += w * vv.z;
    acc.w += w * vv.w;
  }
  float4* p4 = (float4*)(part + (size_t)(b * 8 + c) * DD);
  p4[tid] = acc;
}

// ---------------------------------------------------------------------------
// Kernel 5: out[b,d] = sum_c part[b,c,d]  (fixed-order, deterministic)
// ---------------------------------------------------------------------------
__global__ __launch_bounds__(256)
void sdpa_reduce_kernel(const float* __restrict__ part,
                        float* __restrict__ out) {
  const int idx = blockIdx.x * 256 + threadIdx.x;  // 0..32767
  const int b = idx >> 10;
  const int d = idx & 1023;
  float s = 0.f;
  #pragma unroll
  for (int c = 0; c < 8; ++c)
    s += part[(size_t)(b * 8 + c) * DD + d];
  out[idx] = s;
}

// ---------------------------------------------------------------------------
// Host-side launcher
// inputs: q[32,1024] f32, k[32,4096,1024] f32, v[32,4096,1024] f32,
//         mask[32,4096] bool (dead in reference), W[1024,1024] f32
// d_out:  out[32,1024] then attn[32,4096], both f32, concatenated flat.
// ---------------------------------------------------------------------------
extern "C" void kernel_launch(void* const* d_in, const int* in_sizes, int n_in,
                              void* d_out, int out_size, void* d_ws, size_t ws_size,
                              hipStream_t stream) {
  const float* q = (const float*)d_in[0];
  const float* k = (const float*)d_in[1];
  const float* v = (const float*)d_in[2];
  // d_in[3] = mask : intentionally unused (reference discards masked_fill)
  const float* W = (const float*)d_in[4];

  float* out  = (float*)d_out;            // [32,1024]
  float* attn = out + (size_t)BB * DD;    // [32,4096]

  // workspace layout (all within first 2 MB of d_ws)
  float* qW    = (float*)d_ws;                               // 32*1024  = 128 KB
  float* score = (float*)((char*)d_ws + (size_t)(128 << 10)); // 32*4096  = 512 KB
  float* part  = (float*)((char*)d_ws + (size_t)(640 << 10)); // 32*8*1024 = 1 MB

  // 1) qW = q @ W  (fp32 WMMA)
  sdpa_qw_wmma_kernel<<<dim3(BB / 16, DD / 16), 32, 0, stream>>>(q, W, qW);
  // 2) score[b,s] = k[b,s,:] . qW[b,:]
  sdpa_score_kernel<<<dim3(SS / 128, BB), 256, 0, stream>>>(k, qW, score);
  // 3) attn = softmax(score) -> second output slot
  sdpa_softmax_kernel<<<BB, 256, 0, stream>>>(score, attn);
  // 4) partial out over 8 s-chunks
  sdpa_outpart_kernel<<<dim3(8, BB), 256, 0, stream>>>(v, attn, part);
  // 5) reduce partials -> first output slot
  sdpa_reduce_kernel<<<(BB * DD) / 256, 256, 0, stream>>>(part, out);
}